// VectorQuantizer_25993142075529
// MI455X (gfx1250) — compile-verified
//
#include <hip/hip_runtime.h>

// Problem constants (from reference): F=32, N=4096, D=64, K=512
#define F_ 32
#define N_ 4096
#define D_ 64
#define K_ 512

typedef __attribute__((ext_vector_type(2))) float v2f;
typedef __attribute__((ext_vector_type(4))) float v4f;
typedef __attribute__((ext_vector_type(8))) float v8f;

// LDS layout (floats), relative to dynamic-LDS base (= 0, no static LDS).
// Pad 68 dwords/row: 68 % 64 == 4 -> bank = (4*row + col) % 64:
//  - B-frag reads  (lanes vary k, col pair d,d+1)   : conflict-free b64
//  - A-frag reads  (lanes vary row, col pair d,d+1) : conflict-free b64
//  - output gather (lanes vary d, fixed k)          : conflict-free b64
#define WP 68
#define XP 68
#define LDS_W   0                        // swT[k][d] : 512 x 68
#define LDS_X   (K_ * WP)                // sx [row][d]: 128 x 68
#define LDS_WN  (LDS_X + 128 * XP)       // ||w_k||^2 : 512
#define LDS_SS  (LDS_WN + K_)            // per-wave min scores : 8 x 32
#define LDS_SK  (LDS_SS + 8 * 32)        // per-wave argmin idx : 8 x 32
#define LDS_FLOATS (LDS_SK + 8 * 32)     // total: 44544 floats = 178176 B

__global__ void vq_zero_loss(float* loss) { *loss = 0.0f; }

__global__ __launch_bounds__(256)
void vq_fused_kernel(const float* __restrict__ x,   // [F, N, D]
                     const float* __restrict__ w,   // [F, D, K]
                     float* __restrict__ out,       // [F, N, D]
                     float* __restrict__ loss) {    // scalar accumulator
  extern __shared__ float smem[];
  float* swT = smem + LDS_W;            // transposed w[f]: [K][WP]
  float* sx  = smem + LDS_X;            // x chunk: [128][XP]
  float* swn = smem + LDS_WN;           // ||w_k||^2
  float* sS  = smem + LDS_SS;
  int*   sK  = (int*)(smem + LDS_SK);

  const int f     = blockIdx.x >> 5;    // 32 row-chunks per f
  const int chunk = blockIdx.x & 31;
  const int tid   = threadIdx.x;
  const int wave  = tid >> 5;
  const int lane  = tid & 31;

  // ---- 1) Kick off async DMA of the x chunk (128 x 64 f32, linear) ----
  // global_load_async_to_lds_b128: VDST = LDS byte address, VADDR = 32-bit
  // byte offset, SADDR = 64-bit base (GVS mode). Tracked by ASYNCcnt.
  {
    const float* xbase = x + ((size_t)f * N_ + (size_t)chunk * 128) * D_;
#pragma unroll
    for (int i = 0; i < 8; ++i) {
      int e   = (tid + i * 256) * 4;                 // first of 4 floats
      int row = e >> 6;                              // /64
      int col = e & 63;
      unsigned ldsoff = (unsigned)(LDS_X + row * XP + col) * 4u;  // 16B aligned
      unsigned goff   = (unsigned)e * 4u;
      asm volatile("global_load_async_to_lds_b128 %0, %1, %2"
                   :: "v"(ldsoff), "v"(goff), "s"(xbase)
                   : "memory");
    }
  }

  // ---- 2) Stage w[f] transposed into LDS (overlaps the async x DMA) ----
  {
    const float* wf = w + (size_t)f * (D_ * K_);
    for (int i = tid * 4; i < D_ * K_; i += 256 * 4) {
      v4f v = *(const v4f*)(wf + i);                 // coalesced b128
      int d  = i >> 9;                               // /512
      int k0 = i & (K_ - 1);
#pragma unroll
      for (int j = 0; j < 4; ++j) swT[(k0 + j) * WP + d] = v[j];
    }
  }

  // All async x-loads done (per wave), then block-wide barrier covers both
  // the w transpose stores and everyone's async completions.
  asm volatile("s_wait_asynccnt 0x0" ::: "memory");
  __syncthreads();

  // ---- 3) ||w_k||^2 from transposed layout (contiguous per thread) ----
  for (int k = tid; k < K_; k += 256) {
    const float* row = swT + k * WP;
    float s = 0.0f;
#pragma unroll
    for (int d = 0; d < D_; d += 2) { v2f v = *(const v2f*)(row + d); s += v.x * v.x + v.y * v.y; }
    swn[k] = s;
  }
  __syncthreads();

  // ---- WMMA fragment coordinates (ISA 7.12.2, wave32) ----
  // A (16x4 f32): A[v][lane] = x[row = lane%16, kk = v + 2*(lane/16)]
  // B (4x16 f32): B[v][lane] = w[kk  = v + 2*(lane/16), col = lane%16]
  // C (16x16)   : C[v][lane] = dot[row = v + 8*(lane/16), col = lane%16]
  const int r    = lane & 15;
  const int half = lane >> 4;
  const int c    = half << 1;                 // 0 or 2
  const int n0   = chunk * 128 + wave * 16;   // this wave's 16 global rows

  // ---- 4) A fragments from LDS (b64, conflict-free); x-norm partials ----
  const float* xrow = sx + (wave * 16 + r) * XP;
  v2f afrag[16];
  float xnp = 0.0f;
#pragma unroll
  for (int s = 0; s < 16; ++s) {
    v2f a = *(const v2f*)(xrow + 4 * s + c);
    afrag[s] = a;
    xnp += a.x * a.x + a.y * a.y;
  }
  // low half covers d%4 in {0,1}, high half {2,3}; combine halves:
  float xnorm = xnp + __shfl_xor(xnp, 16, 32);  // lane holds ||x_row||^2, row = lane%16

  float minS[8];
  int   minK[8];
#pragma unroll
  for (int v = 0; v < 8; ++v) { minS[v] = 3.4e38f; minK[v] = 0; }

  // ---- 5) Fused GEMM + running argmin, double-buffered B fragments ----
  // Tile t's 16 B-fragments sit in registers while tile t+1's 8
  // ds_load_2addr_b64 are in flight -> WMMAs overlap LDS latency instead of
  // stalling on s_wait_dscnt 0 per fragment pair.
  v2f b0[16], b1[16];

#define LOAD_B(dst, tile)                                                   \
  {                                                                         \
    const float* brow_ = swT + ((tile) * 16 + r) * WP;                      \
    _Pragma("unroll")                                                       \
    for (int s = 0; s < 16; ++s) dst[s] = *(const v2f*)(brow_ + 4 * s + c); \
  }

#define PROCESS(tile, buf)                                                  \
  {                                                                         \
    const int kcol_ = (tile) * 16 + r;                                      \
    const float wn_ = swn[kcol_];                                           \
    v8f acc = {};                                                           \
    _Pragma("unroll")                                                       \
    for (int s = 0; s < 16; ++s)                                            \
      acc = __builtin_amdgcn_wmma_f32_16x16x4_f32(                          \
          false, afrag[s], false, buf[s], (short)0, acc, false, false);     \
    _Pragma("unroll")                                                       \
    for (int v = 0; v < 8; ++v) {                                           \
      float sv = wn_ - 2.0f * acc[v]; /* ||x||^2 is row-constant */         \
      if (sv < minS[v]) { minS[v] = sv; minK[v] = kcol_; }                  \
    }                                                                       \
  }

  LOAD_B(b0, 0);
  for (int t = 0; t < 32; t += 2) {
    LOAD_B(b1, t + 1);          // prefetch odd tile
    PROCESS(t, b0);             // overlaps with b1 loads in flight
    LOAD_B(b0, (t + 2) & 31);   // prefetch next even tile (wraps harmlessly)
    PROCESS(t + 1, b1);         // overlaps with b0 loads in flight
  }
#undef LOAD_B
#undef PROCESS

  // ---- 6) Cross-lane argmin within each 16-lane half (lowest idx on tie) ----
#pragma unroll
  for (int m = 8; m >= 1; m >>= 1) {
#pragma unroll
    for (int v = 0; v < 8; ++v) {
      float oS = __shfl_xor(minS[v], m, 32);
      int   oK = __shfl_xor(minK[v], m, 32);
      if (oS < minS[v] || (oS == minS[v] && oK < minK[v])) { minS[v] = oS; minK[v] = oK; }
    }
  }

  // lanes 0 and 16 publish rows 0-7 / 8-15 of this wave
  if (r == 0) {
#pragma unroll
    for (int v = 0; v < 8; ++v) {
      sS[wave * 32 + half * 8 + v] = minS[v];
      sK[wave * 32 + half * 8 + v] = minK[v];
    }
  }
  __syncthreads();

  // ---- 7) Loss: sum over this wave's 16 rows of (||x||^2 + minScore) ----
  float dv = (lane < 16) ? (xnorm + sS[wave * 32 + lane]) : 0.0f;
  dv += __shfl_xor(dv, 1, 32);
  dv += __shfl_xor(dv, 2, 32);
  dv += __shfl_xor(dv, 4, 32);
  dv += __shfl_xor(dv, 8, 32);
  if (lane == 0) {
    // loss = (1 + commitment_cost) * sum(minDist) / (F*N*D)
    atomicAdd(loss, dv * (1.25f / ((float)F_ * (float)N_ * (float)D_)));
  }

  // ---- 8) Gather quantized codewords (b64 LDS, coalesced b64 stores) ----
  float* orow = out + ((size_t)f * N_ + n0) * D_;
  for (int row = 0; row < 16; ++row) {
    int k = sK[wave * 32 + row];
    v2f q = *(const v2f*)(swT + k * WP + 2 * lane);    // conflict-free
    *(v2f*)(orow + row * D_ + 2 * lane) = q;           // 256B coalesced
  }
}

extern "C" void kernel_launch(void* const* d_in, const int* in_sizes, int n_in,
                              void* d_out, int out_size, void* d_ws, size_t ws_size,
                              hipStream_t stream) {
  const float* x = (const float*)d_in[0];   // [F, N, D] f32
  const float* w = (const float*)d_in[1];   // [F, D, K] f32
  float* out  = (float*)d_out;              // [F*N*D] output ++ [1] loss
  float* loss = out + (out_size - 1);

  vq_zero_loss<<<1, 1, 0, stream>>>(loss);

  const size_t smem_bytes = (size_t)LDS_FLOATS * sizeof(float);  // ~178 KB < 320 KB WGP LDS
  vq_fused_kernel<<<F_ * 32, 256, smem_bytes, stream>>>(x, w, out, loss);
}